// TransformedReLU_15221364097584
// MI455X (gfx1250) — compile-verified
//
#include <hip/hip_runtime.h>

typedef float v2f __attribute__((ext_vector_type(2)));
typedef float v4f __attribute__((ext_vector_type(4)));
typedef float v8f __attribute__((ext_vector_type(8)));

#define NCOL 8192            // neurons
#define NERR 512             // error terms
#define EOLD 513             // 1 + NERR rows in the input / transformed block
#define NROWS_OUT (EOLD + NCOL)  // 8705 output rows

// ---------------------------------------------------------------------------
// Kernel A: per-column zonotope stats.
// abs_sum[n] = sum_e |x[1+e, n]| computed as ones^T * |X| with
// V_WMMA_F32_16X16X4_F32 (fp32 accumulate -> same precision class as VALU).
// Each wave owns 16 columns; lane L supplies B[k, n=L%16] for two k's.
// ---------------------------------------------------------------------------
__global__ __launch_bounds__(256) void zono_stats_wmma(
    const float* __restrict__ x, float* __restrict__ scale,
    float* __restrict__ center, float* __restrict__ vals,
    int* __restrict__ cross)
{
  const int lane    = threadIdx.x & 31;
  const int wave    = threadIdx.x >> 5;
  const int colBase = blockIdx.x * 128 + wave * 16;
  const int n       = colBase + (lane & 15);
  const int kh      = (lane >> 4) << 1;           // lanes 0-15 -> rows {0,1}, 16-31 -> {2,3}

  v2f ones = {1.0f, 1.0f};
  v8f acc  = {};
  const float* p = x + (size_t)NCOL + n;          // skip center row

  for (int r = 0; r < NERR; r += 4) {
    if (r + 8 < NERR)                              // uniform guard: speculative row prefetch
      __builtin_prefetch(p + (size_t)(r + kh + 8) * NCOL, 0, 0);
    v2f b;
    b[0] = __builtin_fabsf(p[(size_t)(r + kh)     * NCOL]);
    b[1] = __builtin_fabsf(p[(size_t)(r + kh + 1) * NCOL]);
    // D = A(ones) x B(|x| tile) + C  -> every row of D = column abs-sums
    acc = __builtin_amdgcn_wmma_f32_16x16x4_f32(false, ones, false, b,
                                                (short)0, acc, false, false);
  }
  float abs_sum = acc[0];                          // same value in all lanes of a column

  if (lane < 16) {
    float c  = x[n];                               // center row
    float up = c + abs_sum;
    float lo = c - abs_sum;
    int crossb = (lo * up < 0.0f);
    int posb   = (lo >= 0.0f);
    float denom = (up == lo) ? 1.0f : (up - lo);
    float lam   = (float)posb + (crossb ? (up / denom) : 0.0f);
    float delta = fmaxf(-lam * lo, (1.0f - lam) * up);
    float cr = (float)crossb, po = (float)posb;
    center[n] = (delta * 0.5f + lam * c) * cr + c * po;
    scale[n]  = lam * cr + po;
    vals[n]   = delta * 0.5f * cr;
    cross[n]  = crossb;
  }
}

// ---------------------------------------------------------------------------
// Kernel B: 285 MB output writer (bandwidth-critical).
// Row 0: new center. Rows 1..512: x*scale. Rows 513..8704: zeros.
// Non-temporal b128 stores: output (285MB) exceeds the 192MB L2 -> bypass.
// ---------------------------------------------------------------------------
__global__ __launch_bounds__(256) void zono_write(
    const float* __restrict__ x, const float* __restrict__ scale,
    const float* __restrict__ center, float* __restrict__ out)
{
  const int row  = blockIdx.y;
  const int col4 = (blockIdx.x * 256 + threadIdx.x) * 4;   // blockIdx.x in [0,8)
  v4f v;
  if (row >= EOLD) {
    v = (v4f){0.0f, 0.0f, 0.0f, 0.0f};
  } else if (row == 0) {
    v = *(const v4f*)(center + col4);
  } else {
    v4f xv = *(const v4f*)(x + (size_t)row * NCOL + col4);
    v4f sv = *(const v4f*)(scale + col4);
    v = xv * sv;
  }
  __builtin_nontemporal_store(v, (v4f*)(out + (size_t)row * NCOL + col4));
}

// ---------------------------------------------------------------------------
// Kernel C: prefix-sum of crossing flags (single block, LDS scan) + scatter
// of delta/2 into row 513 + rank(n). Runs after the zero-fill in stream order.
// Non-crossing columns add exactly 0 in the reference -> skipping them is
// equivalent.
// ---------------------------------------------------------------------------
__global__ __launch_bounds__(256) void zono_scan_scatter(
    const int* __restrict__ cross, const float* __restrict__ vals,
    float* __restrict__ out)
{
  __shared__ int part[256];
  const int t    = threadIdx.x;
  const int base = t * 32;                         // 256 threads x 32 cols = 8192

  int s = 0;
  for (int i = 0; i < 32; ++i) s += cross[base + i];
  part[t] = s;
  __syncthreads();

  for (int off = 1; off < 256; off <<= 1) {        // Hillis-Steele inclusive scan
    int v = (t >= off) ? part[t - off] : 0;
    __syncthreads();
    part[t] += v;
    __syncthreads();
  }
  int run = part[t] - s;                           // exclusive prefix = rank of first cross

  for (int i = 0; i < 32; ++i) {
    int n = base + i;
    if (cross[n]) {
      out[(size_t)(EOLD + run) * NCOL + n] = vals[n];
      run++;
    }
  }
}

// ---------------------------------------------------------------------------
extern "C" void kernel_launch(void* const* d_in, const int* in_sizes, int n_in,
                              void* d_out, int out_size, void* d_ws, size_t ws_size,
                              hipStream_t stream) {
  const float* x = (const float*)d_in[0];          // [513, 8192] fp32
  float* out = (float*)d_out;                      // [8705, 8192] fp32
  float* ws  = (float*)d_ws;                       // 128 KB scratch
  float* scale  = ws;
  float* center = ws + NCOL;
  float* vals   = ws + 2 * NCOL;
  int*   cross  = (int*)(ws + 3 * NCOL);

  zono_stats_wmma<<<dim3(NCOL / 128), 256, 0, stream>>>(x, scale, center, vals, cross);
  zono_write<<<dim3(NCOL / (256 * 4), NROWS_OUT), 256, 0, stream>>>(x, scale, center, out);
  zono_scan_scatter<<<1, 256, 0, stream>>>(cross, vals, out);
}